// TurbulenceRCLN_44341242363940
// MI455X (gfx1250) — compile-verified
//
#include <hip/hip_runtime.h>
#include <math.h>

// ---------------- problem constants ----------------
#define RES   64
#define BB    16
#define WID   32
#define MM1   12
#define MM2   12
#define NSP   (BB*RES*RES)          // 65536 FNO spatial points
#define N3    (RES*RES*RES)         // 262144
#define DXF   0.09817477042468103f  // 2*pi/64
#define DTF   0.001f
#define NUF   0.002f
#define TWO_PI_OVER_64 0.09817477042468103f

typedef float v2f __attribute__((ext_vector_type(2)));
typedef float v8f __attribute__((ext_vector_type(8)));

__device__ __forceinline__ float gelu_tanh(float x) {
    float x3 = x * x * x;
    return 0.5f * x * (1.0f + tanhf(0.7978845608028654f * (x + 0.044715f * x3)));
}

// ---------------- 1) closure input: grads + strain mag at y=32 plane ----------------
// cin layout: [m][12] (cols 10,11 zero-padded), m = (b*64 + p)*64 + q
__global__ __launch_bounds__(256) void closure_kernel(const float* __restrict__ v,
                                                      float* __restrict__ cin) {
    int t = blockIdx.x * blockDim.x + threadIdx.x;
    if (t >= NSP) return;
    int q = t & 63;
    int p = (t >> 6) & 63;
    int b = t >> 12;
    const float inv2dx = 1.0f / (2.0f * DXF);
    float g[3][3];
#pragma unroll
    for (int i = 0; i < 3; i++) {
        size_t base = ((size_t)(b * 3 + i)) * (size_t)N3;
        int pp = (p + 1) & 63, pm = (p + 63) & 63;
        float uc  = v[base + ((size_t)p  << 12) + (32 << 6) + q];
        float uxp = v[base + ((size_t)pp << 12) + (32 << 6) + q];
        float uxm = v[base + ((size_t)pm << 12) + (32 << 6) + q];
        float uyp = v[base + ((size_t)p  << 12) + (33 << 6) + q];
        float uym = v[base + ((size_t)p  << 12) + (31 << 6) + q];
        g[i][0] = (uxp - uxm) * inv2dx;
        g[i][1] = (uyp - uym) * inv2dx;
        g[i][2] = uc * 0.1f;
    }
    float ss = 0.0f;
#pragma unroll
    for (int i = 0; i < 3; i++)
#pragma unroll
        for (int j = 0; j < 3; j++) {
            float s = 0.5f * (g[i][j] + g[j][i]);
            ss += s * s;
        }
    float* o = cin + (size_t)t * 12;
#pragma unroll
    for (int i = 0; i < 3; i++)
#pragma unroll
        for (int j = 0; j < 3; j++) o[i * 3 + j] = g[i][j];
    o[9]  = sqrtf(2.0f * ss);
    o[10] = 0.0f;
    o[11] = 0.0f;
}

// ---------------- 2) fp32 WMMA GEMM (compile-time shapes, LDS-staged weights) ----------
// Y[M,NPAD] = act( X[M,KP] * W[K,N] + bias (+ S[M,NPAD]) )
// KP = K rounded up to 4 (X buffer must have KP columns); NPAD multiple of 16.
template <int K, int N, int NPAD, bool ADD_S, bool GELU>
__global__ __launch_bounds__(128) void wmma_gemm(const float* __restrict__ X,
                                                 const float* __restrict__ W,
                                                 const float* __restrict__ bias,
                                                 const float* __restrict__ S,
                                                 float* __restrict__ Y, int M) {
    constexpr int KP = (K + 3) & ~3;
    constexpr int LD = NPAD + 1;               // LDS pitch: break bank conflicts
    __shared__ float lw[KP * LD];
    __shared__ float lb[NPAD];
    int tid = threadIdx.y * 32 + threadIdx.x;
    for (int idx = tid; idx < KP * NPAD; idx += 128) {
        int k = idx / NPAD, n = idx - k * NPAD;
        lw[k * LD + n] = (k < K && n < N) ? W[k * N + n] : 0.0f;
    }
    if (tid < NPAD) lb[tid] = (tid < N) ? bias[tid] : 0.0f;
    __syncthreads();

    constexpr int tilesN = NPAD >> 4;
    int tile = blockIdx.x * 4 + threadIdx.y;
    if (tile >= (M >> 4) * tilesN) return;     // wave-uniform
    int tm = tile / tilesN, tn = tile - tm * tilesN;
    int m0 = tm << 4, n0 = tn << 4;
    int lane = threadIdx.x;
    int half = lane >> 4, l = lane & 15;
    int mrow = m0 + l;
    int ncol = n0 + l;
    const float* xr = X + (size_t)mrow * KP;
    __builtin_prefetch(xr, 0, 0);              // global_prefetch_b8
    v8f acc = {};
#pragma unroll
    for (int kk = 0; kk < KP; kk += 4) {
        int k0 = kk + 2 * half;                // lanes 0-15: K=kk,kk+1 ; 16-31: kk+2,kk+3
        v2f a = *reinterpret_cast<const v2f*>(xr + k0);
        v2f b;
        b.x = lw[k0 * LD + ncol];
        b.y = lw[(k0 + 1) * LD + ncol];
        acc = __builtin_amdgcn_wmma_f32_16x16x4_f32(false, a, false, b,
                                                    (short)0, acc, false, false);
    }
    float bv = lb[ncol];
#pragma unroll
    for (int r = 0; r < 8; r++) {
        int m = m0 + half * 8 + r;             // C/D layout: vgpr r -> M=r / M=8+r
        float val = acc[r] + bv;
        if (ADD_S) val += S[(size_t)m * NPAD + ncol];
        if (GELU) val = gelu_tanh(val);
        Y[(size_t)m * NPAD + ncol] = val;
    }
}

// ---------------- 3) spectral conv via truncated direct DFTs ----------------
// F1[b][c][p][kq] = sum_q h[m][c] * e^{-i 2pi kq q /64}, kq in [0,12)
__global__ __launch_bounds__(256) void dft_y_kernel(const float* __restrict__ h,
                                                    float* __restrict__ F1) {
    int t = blockIdx.x * blockDim.x + threadIdx.x;
    if (t >= BB * WID * RES * MM2) return;
    int kq = t % MM2;
    int p  = (t / MM2) % RES;
    int c  = (t / (MM2 * RES)) % WID;
    int b  = t / (MM2 * RES * WID);
    float re = 0.0f, im = 0.0f;
    for (int q = 0; q < RES; q++) {
        float x = h[((size_t)((b * 64 + p) * 64 + q)) * WID + c];
        int ph = (kq * q) & 63;
        float sn, cs; __sincosf(TWO_PI_OVER_64 * ph, &sn, &cs);
        re += x * cs; im -= x * sn;
    }
    F1[2 * (size_t)t] = re; F1[2 * (size_t)t + 1] = im;
}

// Z[b][c][kpi][kq] = sum_p F1 * e^{-i 2pi kp p /64}, kpi 0..23 (kp: 0..11, 52..63)
__global__ __launch_bounds__(256) void dft_x_kernel(const float* __restrict__ F1,
                                                    float* __restrict__ Z) {
    int t = blockIdx.x * blockDim.x + threadIdx.x;
    if (t >= BB * WID * 24 * MM2) return;
    int kq  = t % MM2;
    int kpi = (t / MM2) % 24;
    int c   = (t / (MM2 * 24)) % WID;
    int b   = t / (MM2 * 24 * WID);
    int kp = (kpi < 12) ? kpi : kpi + 40;
    float re = 0.0f, im = 0.0f;
    for (int p = 0; p < RES; p++) {
        size_t fi = (((size_t)(b * WID + c) * RES) + p) * MM2 + kq;
        float ar = F1[2 * fi], ai = F1[2 * fi + 1];
        int ph = (kp * p) & 63;
        float sn, cs; __sincosf(TWO_PI_OVER_64 * ph, &sn, &cs);
        re += ar * cs + ai * sn;
        im += ai * cs - ar * sn;
    }
    Z[2 * (size_t)t] = re; Z[2 * (size_t)t + 1] = im;
}

// Mx[b][o][kpi][kq] = sum_i Z[b][i][kpi][kq] * w[i][o][kpm][kq]
__global__ __launch_bounds__(256) void mode_mix_kernel(const float* __restrict__ Z,
                                                       const float* __restrict__ w1,
                                                       const float* __restrict__ w2,
                                                       float* __restrict__ Mx) {
    int t = blockIdx.x * blockDim.x + threadIdx.x;
    if (t >= BB * WID * 24 * MM2) return;
    int kq  = t % MM2;
    int kpi = (t / MM2) % 24;
    int o   = (t / (MM2 * 24)) % WID;
    int b   = t / (MM2 * 24 * WID);
    const float* wp = (kpi < 12) ? w1 : w2;
    int kpm = (kpi < 12) ? kpi : kpi - 12;
    float re = 0.0f, im = 0.0f;
    for (int i = 0; i < WID; i++) {
        size_t zi = (((size_t)(b * WID + i) * 24) + kpi) * MM2 + kq;
        float zr = Z[2 * zi], zim = Z[2 * zi + 1];
        size_t wi = ((((size_t)i * WID + o) * MM1) + kpm) * MM2 + kq;
        float wr = wp[2 * wi], wim = wp[2 * wi + 1];
        re += zr * wr - zim * wim;
        im += zr * wim + zim * wr;
    }
    Mx[2 * (size_t)t] = re; Mx[2 * (size_t)t + 1] = im;
}

// G[b][o][p][kq] = (1/64) sum_kpi Mx * e^{+i 2pi kp p/64}
__global__ __launch_bounds__(256) void idft_x_kernel(const float* __restrict__ Mx,
                                                     float* __restrict__ G) {
    int t = blockIdx.x * blockDim.x + threadIdx.x;
    if (t >= BB * WID * RES * MM2) return;
    int kq = t % MM2;
    int p  = (t / MM2) % RES;
    int o  = (t / (MM2 * RES)) % WID;
    int b  = t / (MM2 * RES * WID);
    float re = 0.0f, im = 0.0f;
#pragma unroll
    for (int kpi = 0; kpi < 24; kpi++) {
        int kp = (kpi < 12) ? kpi : kpi + 40;
        size_t mi = (((size_t)(b * WID + o) * 24) + kpi) * MM2 + kq;
        float mr = Mx[2 * mi], mim = Mx[2 * mi + 1];
        int ph = (kp * p) & 63;
        float sn, cs; __sincosf(TWO_PI_OVER_64 * ph, &sn, &cs);
        re += mr * cs - mim * sn;
        im += mr * sn + mim * cs;
    }
    G[2 * (size_t)t] = re * (1.0f / 64.0f);
    G[2 * (size_t)t + 1] = im * (1.0f / 64.0f);
}

// irfft along q (Hermitian): s[m][o] = (1/64)(Re G[...,0] + 2*sum_{kq>=1} Re(G e^{i th}))
__global__ __launch_bounds__(256) void idft_y_kernel(const float* __restrict__ G,
                                                     float* __restrict__ Sout) {
    int t = blockIdx.x * blockDim.x + threadIdx.x;
    if (t >= NSP * WID) return;
    int o = t & 31;
    int m = t >> 5;
    int q = m & 63;
    int p = (m >> 6) & 63;
    int b = m >> 12;
    size_t gb = (((size_t)(b * WID + o) * RES) + p) * MM2;
    float acc = G[2 * gb];                    // kq = 0 : real part only
#pragma unroll
    for (int kq = 1; kq < MM2; kq++) {
        float gr = G[2 * (gb + kq)], gi = G[2 * (gb + kq) + 1];
        int ph = (kq * q) & 63;
        float sn, cs; __sincosf(TWO_PI_OVER_64 * ph, &sn, &cs);
        acc += 2.0f * (gr * cs - gi * sn);
    }
    Sout[(size_t)t] = acc * (1.0f / 64.0f);
}

// ---------------- 4) 2-D divergence of tau (z-term vanishes) ----------------
// tau: [m][16] (cols 0..5 valid) ; dv: [b][3][64][64]
__global__ __launch_bounds__(256) void divtau_kernel(const float* __restrict__ tau,
                                                     float* __restrict__ dv) {
    int t = blockIdx.x * blockDim.x + threadIdx.x;
    if (t >= BB * RES * RES) return;
    int q = t & 63, p = (t >> 6) & 63, b = t >> 12;
    const float inv2dx = 1.0f / (2.0f * DXF);
    int pp = (p + 1) & 63, pm = (p + 63) & 63, qp = (q + 1) & 63, qm = (q + 63) & 63;
#define TAU(ch, PP, QQ) tau[((size_t)((b << 12) | ((PP) << 6) | (QQ))) * 16 + (ch)]
    float d0 = (TAU(0, pp, q) - TAU(0, pm, q) + TAU(3, p, qp) - TAU(3, p, qm)) * inv2dx;
    float d1 = (TAU(3, pp, q) - TAU(3, pm, q) + TAU(1, p, qp) - TAU(1, p, qm)) * inv2dx;
    float d2 = (TAU(4, pp, q) - TAU(4, pm, q) + TAU(5, p, qp) - TAU(5, p, qm)) * inv2dx;
#undef TAU
    size_t base = (size_t)b * 3 * 4096 + (size_t)((p << 6) | q);
    dv[base] = d0; dv[base + 4096] = d1; dv[base + 2 * 4096] = d2;
}

// ---------------- 5) fused NS step, vectorized x4 along z ----------------
__global__ __launch_bounds__(256) void ns_kernel(const float* __restrict__ v,
                                                 const float* __restrict__ dv,
                                                 float* __restrict__ out) {
    int t = blockIdx.x * blockDim.x + threadIdx.x;  // B*64*64*16 threads
    if (t >= BB * 64 * 64 * 16) return;
    int zq = t & 15;
    int y  = (t >> 4) & 63;
    int x  = (t >> 10) & 63;
    int b  = t >> 16;
    int z0 = zq << 2;
    const float inv2dx = 1.0f / (2.0f * DXF);
    const float invdx2 = 1.0f / (DXF * DXF);
    int xp = (x + 1) & 63, xm = (x + 63) & 63;
    int yp = (y + 1) & 63, ym = (y + 63) & 63;
    int zmi = (z0 + 63) & 63, zpi = (z0 + 4) & 63;

    float uc[3][4], ddx[3][4], ddy[3][4], ddz[3][4], lap[3][4];
#pragma unroll
    for (int i = 0; i < 3; i++) {
        size_t base = ((size_t)(b * 3 + i)) << 18;
        size_t o00 = base + ((size_t)x << 12) + (y << 6);
        float4 c4  = *reinterpret_cast<const float4*>(v + o00 + z0);
        float4 xp4 = *reinterpret_cast<const float4*>(v + base + ((size_t)xp << 12) + (y << 6) + z0);
        float4 xm4 = *reinterpret_cast<const float4*>(v + base + ((size_t)xm << 12) + (y << 6) + z0);
        float4 yp4 = *reinterpret_cast<const float4*>(v + base + ((size_t)x << 12) + (yp << 6) + z0);
        float4 ym4 = *reinterpret_cast<const float4*>(v + base + ((size_t)x << 12) + (ym << 6) + z0);
        float zm = v[o00 + zmi];
        float zp = v[o00 + zpi];
        float cz[6]  = { zm, c4.x, c4.y, c4.z, c4.w, zp };
        float axp[4] = { xp4.x, xp4.y, xp4.z, xp4.w };
        float axm[4] = { xm4.x, xm4.y, xm4.z, xm4.w };
        float ayp[4] = { yp4.x, yp4.y, yp4.z, yp4.w };
        float aym[4] = { ym4.x, ym4.y, ym4.z, ym4.w };
#pragma unroll
        for (int e = 0; e < 4; e++) {
            float c = cz[e + 1];
            uc[i][e]  = c;
            ddx[i][e] = (axp[e] - axm[e]) * inv2dx;
            ddy[i][e] = (ayp[e] - aym[e]) * inv2dx;
            ddz[i][e] = (cz[e + 2] - cz[e]) * inv2dx;
            lap[i][e] = (axp[e] + axm[e] + ayp[e] + aym[e] + cz[e + 2] + cz[e]
                         - 6.0f * c) * invdx2;
        }
    }
    size_t sp2d = (size_t)((x << 6) | y);
    float dt0 = dv[((size_t)b * 3 + 0) * 4096 + sp2d];
    float dt1 = dv[((size_t)b * 3 + 1) * 4096 + sp2d];
    float dt2 = dv[((size_t)b * 3 + 2) * 4096 + sp2d];
    size_t voff = ((size_t)x << 12) + (y << 6) + z0;

    float4 r0, r1, r2;
    float t0[4], t1[4], t2[4];
#pragma unroll
    for (int e = 0; e < 4; e++) {
        float c0 = uc[0][e] * ddx[0][e] + uc[1][e] * ddx[1][e] + uc[2][e] * ddx[2][e];
        float c1 = uc[0][e] * ddy[0][e] + uc[1][e] * ddy[1][e] + uc[2][e] * ddy[2][e];
        float c2 = uc[0][e] * ddz[0][e] + uc[1][e] * ddz[1][e] + uc[2][e] * ddz[2][e];
        t0[e] = uc[0][e] + DTF * (-c0 + NUF * lap[0][e] + dt0);
        t1[e] = uc[1][e] + DTF * (-c1 + NUF * lap[1][e] + dt1);
        t2[e] = uc[2][e] + DTF * (-c2 + NUF * lap[2][e] + dt2);
    }
    r0 = make_float4(t0[0], t0[1], t0[2], t0[3]);
    r1 = make_float4(t1[0], t1[1], t1[2], t1[3]);
    r2 = make_float4(t2[0], t2[1], t2[2], t2[3]);
    *reinterpret_cast<float4*>(out + (((size_t)(b * 3 + 0)) << 18) + voff) = r0;
    *reinterpret_cast<float4*>(out + (((size_t)(b * 3 + 1)) << 18) + voff) = r1;
    *reinterpret_cast<float4*>(out + (((size_t)(b * 3 + 2)) << 18) + voff) = r2;
}

// ---------------- host side ----------------
extern "C" void kernel_launch(void* const* d_in, const int* in_sizes, int n_in,
                              void* d_out, int out_size, void* d_ws, size_t ws_size,
                              hipStream_t stream) {
    const float* v     = (const float*)d_in[0];
    const float* fc0_w = (const float*)d_in[1];
    const float* fc0_b = (const float*)d_in[2];
    const float* sw1   = (const float*)d_in[3];
    const float* sw2   = (const float*)d_in[4];
    const float* cw    = (const float*)d_in[5];
    const float* cb    = (const float*)d_in[6];
    const float* fc1_w = (const float*)d_in[7];
    const float* fc1_b = (const float*)d_in[8];
    const float* fc2_w = (const float*)d_in[9];
    const float* fc2_b = (const float*)d_in[10];
    float* out = (float*)d_out;
    float* ws  = (float*)d_ws;

    // workspace layout (floats)
    const size_t o_cin = 0;                          // NSP*12
    const size_t o_ha  = o_cin + (size_t)NSP * 12;   // NSP*32
    const size_t o_hb  = o_ha  + (size_t)NSP * WID;  // NSP*32
    const size_t o_s   = o_hb  + (size_t)NSP * WID;  // NSP*32
    const size_t o_F1  = o_s   + (size_t)NSP * WID;  // B*32*64*12*2
    const size_t o_Z   = o_F1  + (size_t)BB * WID * RES * MM2 * 2;
    const size_t o_Mx  = o_Z   + (size_t)BB * WID * 24 * MM2 * 2;
    const size_t o_G   = o_Mx  + (size_t)BB * WID * 24 * MM2 * 2;
    const size_t o_t1  = o_G   + (size_t)BB * WID * RES * MM2 * 2;   // NSP*128
    const size_t o_tau = o_t1  + (size_t)NSP * 128;                  // NSP*16
    const size_t o_dv  = o_tau + (size_t)NSP * 16;                   // B*3*4096

    float* cin  = ws + o_cin;
    float* ha   = ws + o_ha;
    float* hb   = ws + o_hb;
    float* sbuf = ws + o_s;
    float* F1   = ws + o_F1;
    float* Z    = ws + o_Z;
    float* Mx   = ws + o_Mx;
    float* G    = ws + o_G;
    float* t1   = ws + o_t1;
    float* tau  = ws + o_tau;
    float* dvt  = ws + o_dv;

    dim3 gblk(32, 4);

    // 1) closure input
    closure_kernel<<<(NSP + 255) / 256, 256, 0, stream>>>(v, cin);

    // 2) lift: fc0, K=10 (KP=12), N=32
    wmma_gemm<10, 32, 32, false, false>
        <<<(NSP / 16) * 2 / 4, gblk, 0, stream>>>(cin, fc0_w, fc0_b, nullptr, ha, NSP);

    // 3) four FNO layers
    float* hin = ha;
    float* hout = hb;
    const int nF = BB * WID * RES * MM2;   // 393216
    const int nZ = BB * WID * 24 * MM2;    // 147456
    for (int l = 0; l < 4; l++) {
        dft_y_kernel<<<(nF + 255) / 256, 256, 0, stream>>>(hin, F1);
        dft_x_kernel<<<(nZ + 255) / 256, 256, 0, stream>>>(F1, Z);
        mode_mix_kernel<<<(nZ + 255) / 256, 256, 0, stream>>>(
            Z, sw1 + (size_t)l * WID * WID * MM1 * MM2 * 2,
               sw2 + (size_t)l * WID * WID * MM1 * MM2 * 2, Mx);
        idft_x_kernel<<<(nF + 255) / 256, 256, 0, stream>>>(Mx, G);
        idft_y_kernel<<<(NSP * WID + 255) / 256, 256, 0, stream>>>(G, sbuf);
        const float* wl = cw + (size_t)l * WID * WID;
        const float* bl = cb + (size_t)l * WID;
        if (l < 3)
            wmma_gemm<32, 32, 32, true, true>
                <<<(NSP / 16) * 2 / 4, gblk, 0, stream>>>(hin, wl, bl, sbuf, hout, NSP);
        else
            wmma_gemm<32, 32, 32, true, false>
                <<<(NSP / 16) * 2 / 4, gblk, 0, stream>>>(hin, wl, bl, sbuf, hout, NSP);
        float* tmp = hin; hin = hout; hout = tmp;
    }

    // 4) projection: fc1 (32->128, gelu), fc2 (128->6 padded to 16)
    wmma_gemm<32, 128, 128, false, true>
        <<<(NSP / 16) * 8 / 4, gblk, 0, stream>>>(hin, fc1_w, fc1_b, nullptr, t1, NSP);
    wmma_gemm<128, 6, 16, false, false>
        <<<(NSP / 16) * 1 / 4, gblk, 0, stream>>>(t1, fc2_w, fc2_b, nullptr, tau, NSP);

    // 5) 2-D div tau
    divtau_kernel<<<(BB * RES * RES + 255) / 256, 256, 0, stream>>>(tau, dvt);

    // 6) fused NS update (HBM-bound main kernel)
    ns_kernel<<<(BB * 64 * 64 * 16 + 255) / 256, 256, 0, stream>>>(v, dvt, out);
}